// ACM_GCN_Framework_91156385890843
// MI455X (gfx1250) — compile-verified
//
#include <hip/hip_runtime.h>
#include <hip/hip_bf16.h>

// ---------------------------------------------------------------------------
// ACM-GCN forward for MI455X (gfx1250, wave32).
//  - 3 fp32 GEMMs on the matrix pipe via V_WMMA_F32_16X16X4_F32
//    (A fragments register-resident; B fragments as contiguous ds_load_b64
//     from an XOR-bank-swizzled K-pair-interleaved LDS image of W^T)
//  - SpMM aggregation with per-edge wave32 + f32 global atomics (L2-resident)
//  - fused ReLU / gate / combine epilogue with wave32 shuffle reductions
// ---------------------------------------------------------------------------

typedef float v2f __attribute__((ext_vector_type(2)));
typedef float v8f __attribute__((ext_vector_type(8)));

#define D_FEAT 128

// ---------------- degree / normalization ----------------------------------

__global__ __launch_bounds__(256) void acm_deg_init(float* __restrict__ deg, int n) {
    int i = blockIdx.x * 256 + threadIdx.x;
    if (i < n) deg[i] = 1.0f;                      // self-loop contributes 1
}

__global__ __launch_bounds__(256) void acm_deg_accum(const int* __restrict__ coli,
                                                     float* __restrict__ deg, int e) {
    int i = blockIdx.x * 256 + threadIdx.x;
    if (i < e) atomicAdd(&deg[coli[i]], 1.0f);
}

__global__ __launch_bounds__(256) void acm_dinv(float* __restrict__ deg, int n) {
    int i = blockIdx.x * 256 + threadIdx.x;
    if (i < n) deg[i] = rsqrtf(deg[i]);            // deg >= 1 always
}

// ---------------- 3x GEMM: h = x @ W.T + b  (fp32 WMMA) --------------------
// grid = (ceil((N/16)/8), 3), block = 256 (8 waves). Each wave computes a
// 16-row x 128-col tile of h[blockIdx.y].
//
// LDS layout of B = W^T, K-pair interleaved + XOR bank swizzle:
//   element B[k][n] = W[n][k] stored at dword
//     (k>>1)*256 + ((n*2 + (k&1)) ^ ((k>>1 & 1) << 5))
// so a lane's B fragment {B[2kp][n], B[2kp+1][n]} is one contiguous b64, and
// the two half-waves (kp even / kp odd) hit disjoint 32-bank groups.

__global__ __launch_bounds__(256) void acm_gemm3(
    const float* __restrict__ x,
    const float* __restrict__ W0, const float* __restrict__ W1, const float* __restrict__ W2,
    const float* __restrict__ b0, const float* __restrict__ b1, const float* __restrict__ b2,
    float* __restrict__ h0, float* __restrict__ h1, float* __restrict__ h2,
    int N)
{
    __shared__ float wt[D_FEAT * D_FEAT];          // exactly 64 KB

    const int sel = blockIdx.y;
    const float* __restrict__ W = sel == 0 ? W0 : (sel == 1 ? W1 : W2);
    const float* __restrict__ b = sel == 0 ? b0 : (sel == 1 ? b1 : b2);
    float* __restrict__ h       = sel == 0 ? h0 : (sel == 1 ? h1 : h2);

    const int tid = threadIdx.x;
    // cooperative fill (coalesced global reads; one-time LDS write conflicts ok)
    for (int e = tid; e < D_FEAT * D_FEAT; e += 256) {
        int n  = e >> 7, k = e & 127;
        int kp = k >> 1;
        int off = ((n << 1) + (k & 1)) ^ ((kp & 1) << 5);
        wt[kp * 256 + off] = W[e];
    }
    __syncthreads();

    const int wave = tid >> 5;
    const int lane = tid & 31;
    const int tile = blockIdx.x * 8 + wave;
    if (tile * 16 >= N) return;                    // wave-uniform exit: EXEC stays all-1s

    const int row0 = tile * 16;
    const int m    = lane & 15;                    // M index (A) / N index (B,D)
    const int hi16 = lane >> 4;                    // 0: lanes 0-15, 1: lanes 16-31
    const int kl   = hi16 << 1;                    // K sub-pair base: 0 or 2
    const int mb   = m << 1;                       // dword offset of n-pair in LDS row

    // ---- preload entire per-lane A slice into registers (32 x b64 loads) ----
    const float* __restrict__ arow = x + (size_t)(row0 + m) * D_FEAT + kl;
    v2f areg[32];
    #pragma unroll
    for (int kt = 0; kt < 32; ++kt)
        areg[kt] = *(const v2f*)(arow + (kt << 2));

    v8f acc[8];
    #pragma unroll
    for (int nt = 0; nt < 8; ++nt) acc[nt] = (v8f){0.f,0.f,0.f,0.f,0.f,0.f,0.f,0.f};

    // ---- main loop: pure LDS b64 reads + WMMA ----
    #pragma unroll
    for (int kt = 0; kt < 32; ++kt) {
        const float* __restrict__ wp = &wt[((kt << 1) + hi16) * 256 + mb];
        #pragma unroll
        for (int nt = 0; nt < 8; ++nt) {
            v2f bb = *(const v2f*)(wp + ((nt ^ hi16) << 5));
            acc[nt] = __builtin_amdgcn_wmma_f32_16x16x4_f32(
                false, areg[kt], false, bb, (short)0, acc[nt], false, false);
        }
    }

    // D layout: VGPR v holds M = (lane<16 ? v : 8+v), N = nt*16 + (lane&15)
    const int mo = hi16 << 3;
    #pragma unroll
    for (int nt = 0; nt < 8; ++nt) {
        const int   n    = nt * 16 + m;
        const float bias = b[n];
        float* __restrict__ hp = h + (size_t)(row0 + mo) * D_FEAT + n;
        #pragma unroll
        for (int v = 0; v < 8; ++v)
            hp[(size_t)v * D_FEAT] = acc[nt][v] + bias;
    }
}

// ---------------- aggregation: agg = D^-1/2 A_hat D^-1/2 @ h ----------------

__global__ __launch_bounds__(256) void acm_agg_init(
    const float* __restrict__ dinv,
    const float* __restrict__ hhp, const float* __restrict__ hlp,
    float* __restrict__ aghp, float* __restrict__ aglp, long total)
{
    long idx = (long)blockIdx.x * 256 + threadIdx.x;
    if (idx >= total) return;
    int i   = (int)(idx >> 7);
    float d = dinv[i];
    float w = d * d;                               // self-loop weight
    aghp[idx] = w * hhp[idx];
    aglp[idx] = w * hlp[idx];
}

__global__ __launch_bounds__(256) void acm_spmm_edges(
    const int* __restrict__ rowi, const int* __restrict__ coli,
    const float* __restrict__ dinv,
    const float* __restrict__ hhp, const float* __restrict__ hlp,
    float* __restrict__ aghp, float* __restrict__ aglp, int E)
{
    const int e = blockIdx.x * 8 + (threadIdx.x >> 5);   // one wave32 per edge
    if (e >= E) return;
    const int lane = threadIdx.x & 31;
    const int r = rowi[e];
    const int c = coli[e];
    const float w = dinv[r] * dinv[c];
    const size_t ro = (size_t)r * D_FEAT + lane * 4;
    const size_t co = (size_t)c * D_FEAT + lane * 4;
    float4 vh = *(const float4*)(hhp + ro);
    float4 vl = *(const float4*)(hlp + ro);
    atomicAdd(aghp + co + 0, w * vh.x);
    atomicAdd(aghp + co + 1, w * vh.y);
    atomicAdd(aghp + co + 2, w * vh.z);
    atomicAdd(aghp + co + 3, w * vh.w);
    atomicAdd(aglp + co + 0, w * vl.x);
    atomicAdd(aglp + co + 1, w * vl.y);
    atomicAdd(aglp + co + 2, w * vl.z);
    atomicAdd(aglp + co + 3, w * vl.w);
}

// ---------------- epilogue: ReLU, gates, combine ---------------------------

__device__ __forceinline__ float acm_sigmoid(float v) {
    return 1.0f / (1.0f + __expf(-v));
}

__global__ __launch_bounds__(256) void acm_finalize(
    const float* __restrict__ hhp, const float* __restrict__ aghp,
    const float* __restrict__ aglp, const float* __restrict__ hi,
    const float* __restrict__ wlh, const float* __restrict__ blh,
    const float* __restrict__ wll, const float* __restrict__ bll,
    const float* __restrict__ wli, const float* __restrict__ bli,
    float* __restrict__ out, int N)
{
    const int i = blockIdx.x * 8 + (threadIdx.x >> 5);   // one wave32 per node
    if (i >= N) return;
    const int lane = threadIdx.x & 31;
    const size_t o = (size_t)i * D_FEAT + lane * 4;

    float4 a  = *(const float4*)(hhp + o);
    float4 g  = *(const float4*)(aghp + o);
    float Hh[4] = { fmaxf(a.x - g.x, 0.f), fmaxf(a.y - g.y, 0.f),
                    fmaxf(a.z - g.z, 0.f), fmaxf(a.w - g.w, 0.f) };
    float4 l4 = *(const float4*)(aglp + o);
    float Hl[4] = { fmaxf(l4.x, 0.f), fmaxf(l4.y, 0.f),
                    fmaxf(l4.z, 0.f), fmaxf(l4.w, 0.f) };
    float4 i4 = *(const float4*)(hi + o);
    float Hi[4] = { fmaxf(i4.x, 0.f), fmaxf(i4.y, 0.f),
                    fmaxf(i4.z, 0.f), fmaxf(i4.w, 0.f) };

    float4 wh = *(const float4*)(wlh + lane * 4);
    float4 wl = *(const float4*)(wll + lane * 4);
    float4 wi = *(const float4*)(wli + lane * 4);
    float dh = Hh[0]*wh.x + Hh[1]*wh.y + Hh[2]*wh.z + Hh[3]*wh.w;
    float dl = Hl[0]*wl.x + Hl[1]*wl.y + Hl[2]*wl.z + Hl[3]*wl.w;
    float di = Hi[0]*wi.x + Hi[1]*wi.y + Hi[2]*wi.z + Hi[3]*wi.w;

    #pragma unroll
    for (int off = 16; off > 0; off >>= 1) {       // wave32 tree reduction
        dh += __shfl_xor(dh, off, 32);
        dl += __shfl_xor(dl, off, 32);
        di += __shfl_xor(di, off, 32);
    }

    const float ah = acm_sigmoid(dh + blh[0]);
    const float al = acm_sigmoid(dl + bll[0]);
    const float ai = acm_sigmoid(di + bli[0]);

    float4 r;
    r.x = ah * Hh[0] + al * Hl[0] + ai * Hi[0];
    r.y = ah * Hh[1] + al * Hl[1] + ai * Hi[1];
    r.z = ah * Hh[2] + al * Hl[2] + ai * Hi[2];
    r.w = ah * Hh[3] + al * Hl[3] + ai * Hi[3];
    *(float4*)(out + o) = r;
}

// ---------------- launch ---------------------------------------------------

extern "C" void kernel_launch(void* const* d_in, const int* in_sizes, int n_in,
                              void* d_out, int out_size, void* d_ws, size_t ws_size,
                              hipStream_t stream)
{
    const float* x   = (const float*)d_in[0];
    const int*   ei  = (const int*)d_in[1];
    const float* Whp = (const float*)d_in[2];
    const float* bhp = (const float*)d_in[3];
    const float* Wlp = (const float*)d_in[4];
    const float* blp = (const float*)d_in[5];
    const float* Wi  = (const float*)d_in[6];
    const float* bi  = (const float*)d_in[7];
    const float* wlh = (const float*)d_in[8];
    const float* blh = (const float*)d_in[9];
    const float* wll = (const float*)d_in[10];
    const float* bll = (const float*)d_in[11];
    const float* wli = (const float*)d_in[12];
    const float* bli = (const float*)d_in[13];
    float* out = (float*)d_out;

    const int N = in_sizes[0] / D_FEAT;            // 50000 (multiple of 16)
    const int E = in_sizes[1] / 2;                 // 800000
    const int* rowi = ei;
    const int* coli = ei + E;

    // workspace layout (floats): dinv | h_hp | h_lp | h_i | agg_hp | agg_lp
    float* ws = (float*)d_ws;
    const size_t NP  = (size_t)N;
    const size_t Nal = (NP + 255) & ~(size_t)255;
    float* deg  = ws;                              // becomes dinv in place
    float* hhp  = ws + Nal;
    float* hlp  = hhp + NP * D_FEAT;
    float* hi   = hlp + NP * D_FEAT;
    float* aghp = hi  + NP * D_FEAT;
    float* aglp = aghp + NP * D_FEAT;

    // 1) symmetric GCN normalization weights
    acm_deg_init <<<(N + 255) / 256, 256, 0, stream>>>(deg, N);
    acm_deg_accum<<<(E + 255) / 256, 256, 0, stream>>>(coli, deg, E);
    acm_dinv     <<<(N + 255) / 256, 256, 0, stream>>>(deg, N);

    // 2) three fp32 WMMA GEMMs (h = x @ W.T + b)
    const int tiles = N / 16;
    dim3 gg((tiles + 7) / 8, 3);
    acm_gemm3<<<gg, 256, 0, stream>>>(x, Whp, Wlp, Wi, bhp, blp, bi,
                                      hhp, hlp, hi, N);

    // 3) aggregation (self-loop init + edge scatter-add, hp & lp together)
    const long total = (long)NP * D_FEAT;
    acm_agg_init  <<<(int)((total + 255) / 256), 256, 0, stream>>>(deg, hhp, hlp,
                                                                   aghp, aglp, total);
    acm_spmm_edges<<<(E + 7) / 8, 256, 0, stream>>>(rowi, coli, deg,
                                                    hhp, hlp, aghp, aglp, E);

    // 4) fused epilogue
    acm_finalize<<<(N + 7) / 8, 256, 0, stream>>>(hhp, aghp, aglp, hi,
                                                  wlh, blh, wll, bll, wli, bli,
                                                  out, N);
}